// VectorQuantizer_70755291234679
// MI455X (gfx1250) — compile-verified
//
#include <hip/hip_runtime.h>
#include <hip/hip_bf16.h>

typedef __attribute__((ext_vector_type(2))) float v2f;
typedef __attribute__((ext_vector_type(8))) float v8f;

#define N_TOK 16384
#define N_EMB 16384
#define CDIM  256
#define BDIM  16
#define HW    1024
#define WDIM  32
#define LSTR  260           // padded LDS row stride (floats) -> conflict-free frag reads

// LDS layout for the phase kernels (dynamic LDS starts at offset 0: no statics here)
#define ZB_OFF  0u                          // 64 x LSTR resident block
#define EB0_OFF ((unsigned)(64 * LSTR * 4)) // streamed tile, buffer 0 (16 x LSTR)
#define EB1_OFF (EB0_OFF + (unsigned)(16 * LSTR * 4))
#define PHASE_LDS_BYTES ((size_t)(64 + 16 + 16) * LSTR * sizeof(float))

// ---------------------------------------------------------------------------
// Async DMA of one 16x256 row-major tile into a padded 16xLSTR LDS buffer.
// 128 threads, each lane issues 8x GLOBAL_LOAD_ASYNC_TO_LDS_B128 (16B/lane).
// Tracked by ASYNCcnt; completion order is in-order -> s_wait_asynccnt N works
// as a double-buffer fence.
// ---------------------------------------------------------------------------
__device__ __forceinline__ void async_tile_load(const float* __restrict__ gsrc,
                                                unsigned ldsByteBase, int tid) {
    #pragma unroll
    for (int j = 0; j < 8; ++j) {
        int flat4 = j * 128 + tid;               // float4 index within 16x256 tile
        int row   = flat4 >> 6;                  // 64 float4 per row
        int col4  = flat4 & 63;
        unsigned gOff = (unsigned)((row * CDIM + col4 * 4) * 4);
        unsigned lOff = ldsByteBase + (unsigned)((row * LSTR + col4 * 4) * 4);
        asm volatile("global_load_async_to_lds_b128 %0, %1, %2"
                     :: "v"(lOff), "v"(gOff), "s"(gsrc) : "memory");
    }
}
__device__ __forceinline__ void wait_async_le8()  { asm volatile("s_wait_asynccnt 0x8" ::: "memory"); }
__device__ __forceinline__ void wait_async_zero() { asm volatile("s_wait_asynccnt 0x0" ::: "memory"); }

// ---------------------------------------------------------------------------
// Kernel 1: L2-normalize tokens (gathered from [B,C,H,W]) and embeddings.
// One wave per row; 8 waves / 256-thread block.
// ---------------------------------------------------------------------------
__global__ __launch_bounds__(256) void k_normalize(const float* __restrict__ z,
                                                   const float* __restrict__ emb,
                                                   float* __restrict__ zf,
                                                   float* __restrict__ embN) {
    int wave = threadIdx.x >> 5, lane = threadIdx.x & 31;
    int R = blockIdx.x * 8 + wave;                 // 0..32767 (tokens then codes)
    float vals[8];
    if (R < N_TOK) {
        int b = R >> 10, hw = R & 1023;
        #pragma unroll
        for (int j = 0; j < 8; ++j) {
            int c = lane + 32 * j;
            vals[j] = z[(b * CDIM + c) * HW + hw];
        }
    } else {
        int k = R - N_TOK;
        #pragma unroll
        for (int j = 0; j < 8; ++j) vals[j] = emb[k * CDIM + lane + 32 * j];
    }
    float ss = 0.f;
    #pragma unroll
    for (int j = 0; j < 8; ++j) ss += vals[j] * vals[j];
    #pragma unroll
    for (int off = 16; off > 0; off >>= 1) ss += __shfl_xor(ss, off);
    float inv = 1.0f / fmaxf(sqrtf(ss), 1e-12f);   // .clip(1e-12) semantics
    float* dst = (R < N_TOK) ? (zf + R * CDIM) : (embN + (R - N_TOK) * CDIM);
    #pragma unroll
    for (int j = 0; j < 8; ++j) dst[lane + 32 * j] = vals[j] * inv;
}

// ---------------------------------------------------------------------------
// Fragment convention for V_WMMA_F32_16X16X4_F32 (wave32):
//   A (16x4): lane L, vgpr v -> M = L&15, K = 2*(L>>4) + v
//   B (4x16): lane L, vgpr v -> N = L&15, K = 2*(L>>4) + v   (B[k][n] = rowmajor[n][k])
//   C/D     : lane L, vgpr v -> N = L&15, M = v + 8*(L>>4)
// ---------------------------------------------------------------------------

// Phase 1: token-major. Each 128-thread block owns 64 tokens (4 waves x 16 rows),
// streams all 1024 code tiles via async double-buffered LDS staging.
__global__ __launch_bounds__(128) void k_phase1(const float* __restrict__ zf,
                                                const float* __restrict__ embN,
                                                float* __restrict__ rowM,
                                                float* __restrict__ rowS1,
                                                float* __restrict__ rowS2,
                                                float* __restrict__ rowBest,
                                                int*   __restrict__ rowIdx) {
    extern __shared__ float lds[];
    float* zB = lds;                 // 64 x LSTR  (token rows, resident)
    int tid = threadIdx.x;
    int wave = tid >> 5, lane = tid & 31;
    int r = lane & 15, hi = lane >> 4;
    int rowBase = blockIdx.x * 64;

    // stage 64x256 token rows (once)
    {
        int row = tid >> 1, half = tid & 1;
        const float4* src = (const float4*)(zf + (rowBase + row) * CDIM + half * 128);
        float* dst = zB + row * LSTR + half * 128;
        #pragma unroll
        for (int j = 0; j < 32; ++j) {
            float4 v = src[j];
            dst[4*j+0]=v.x; dst[4*j+1]=v.y; dst[4*j+2]=v.z; dst[4*j+3]=v.w;
        }
    }
    // prefetch first code tile while we finish staging / loading A frags
    async_tile_load(embN, EB0_OFF, tid);
    __syncthreads();

    // cache this wave's A fragments (16 rows x K=256 -> 64 x v2f in VGPRs)
    v2f aF[64];
    {
        const float* arow = zB + (wave * 16 + r) * LSTR + 2 * hi;
        #pragma unroll
        for (int t = 0; t < 64; ++t) { aF[t].x = arow[4*t]; aF[t].y = arow[4*t+1]; }
    }

    float m[8], S1[8], S2[8], best[8]; int bidx[8];
    #pragma unroll
    for (int v = 0; v < 8; ++v) { m[v] = -1.0e9f; S1[v] = 0.f; S2[v] = 0.f; best[v] = -2.f; bidx[v] = 0; }

    for (int tile = 0; tile < N_EMB / 16; ++tile) {
        unsigned curOff = (tile & 1) ? EB1_OFF : EB0_OFF;
        if (tile + 1 < N_EMB / 16) {
            unsigned nxtOff = ((tile + 1) & 1) ? EB1_OFF : EB0_OFF;
            async_tile_load(embN + (size_t)(tile + 1) * 16 * CDIM, nxtOff, tid);
            wait_async_le8();                // oldest 8 (current tile) complete
        } else {
            wait_async_zero();
        }
        __syncthreads();                     // all waves' DMA portions visible

        v8f acc = {0.f,0.f,0.f,0.f,0.f,0.f,0.f,0.f};
        const float* brow = lds + curOff / 4 + r * LSTR + 2 * hi;
        #pragma unroll
        for (int t = 0; t < 64; ++t) {       // K = 256 in steps of 4
            v2f b; b.x = brow[4*t]; b.y = brow[4*t+1];
            acc = __builtin_amdgcn_wmma_f32_16x16x4_f32(false, aF[t], false, b,
                                                        (short)0, acc, false, false);
        }

        int code = tile * 16 + r;            // this lane's column (code id)
        #pragma unroll
        for (int v = 0; v < 8; ++v) {
            float dot = acc[v];
            float s  = 200.0f * dot;         // affinity/T up to a constant shift
            float nm = fmaxf(m[v], s);
            float sc = __expf(m[v] - nm);
            float e  = __expf(s - nm);
            S2[v] = sc * (S2[v] + (m[v] - nm) * S1[v]) + (s - nm) * e;
            S1[v] = sc * S1[v] + e;
            m[v]  = nm;
            if (dot > best[v]) { best[v] = dot; bidx[v] = code; }
        }
        __syncthreads();                     // done reading cur before it is re-filled
    }

    // merge across the 16 lanes of each half-wave (each half holds the same 8 rows)
    #pragma unroll
    for (int off = 1; off < 16; off <<= 1) {
        #pragma unroll
        for (int v = 0; v < 8; ++v) {
            float om  = __shfl_xor(m[v],    off);
            float o1  = __shfl_xor(S1[v],   off);
            float o2  = __shfl_xor(S2[v],   off);
            float ob  = __shfl_xor(best[v], off);
            int   obi = __shfl_xor(bidx[v], off);
            float nm = fmaxf(m[v], om);
            float sA = __expf(m[v] - nm), sB = __expf(om - nm);
            S2[v] = sA * (S2[v] + (m[v] - nm) * S1[v]) + sB * (o2 + (om - nm) * o1);
            S1[v] = sA * S1[v] + sB * o1;
            m[v]  = nm;
            if (ob > best[v] || (ob == best[v] && obi < bidx[v])) { best[v] = ob; bidx[v] = obi; }
        }
    }
    if (r == 0) {
        #pragma unroll
        for (int v = 0; v < 8; ++v) {
            int row = rowBase + wave * 16 + v + 8 * hi;
            rowM[row] = m[v]; rowS1[row] = S1[v]; rowS2[row] = S2[v];
            rowBest[row] = best[v]; rowIdx[row] = bidx[v];
        }
    }
}

// Phase 2: code-major. Each block owns 64 codes, streams all 1024 token tiles,
// accumulates sum_n p[n,k] per code in registers -> avg_probs (no atomics).
__global__ __launch_bounds__(128) void k_phase2(const float* __restrict__ zf,
                                                const float* __restrict__ embN,
                                                const float* __restrict__ rowM,
                                                const float* __restrict__ rowS1,
                                                float* __restrict__ avgProbs) {
    extern __shared__ float lds[];
    float* eBlk = lds;               // 64 x LSTR  (code rows, resident)
    int tid = threadIdx.x;
    int wave = tid >> 5, lane = tid & 31;
    int r = lane & 15, hi = lane >> 4;
    int codeBase = blockIdx.x * 64;

    {
        int row = tid >> 1, half = tid & 1;
        const float4* src = (const float4*)(embN + (codeBase + row) * CDIM + half * 128);
        float* dst = eBlk + row * LSTR + half * 128;
        #pragma unroll
        for (int j = 0; j < 32; ++j) {
            float4 v = src[j];
            dst[4*j+0]=v.x; dst[4*j+1]=v.y; dst[4*j+2]=v.z; dst[4*j+3]=v.w;
        }
    }
    async_tile_load(zf, EB0_OFF, tid);
    __syncthreads();

    v2f aF[64];
    {
        const float* arow = eBlk + (wave * 16 + r) * LSTR + 2 * hi;
        #pragma unroll
        for (int t = 0; t < 64; ++t) { aF[t].x = arow[4*t]; aF[t].y = arow[4*t+1]; }
    }

    float sumP[8];
    #pragma unroll
    for (int v = 0; v < 8; ++v) sumP[v] = 0.f;

    for (int tile = 0; tile < N_TOK / 16; ++tile) {
        unsigned curOff = (tile & 1) ? EB1_OFF : EB0_OFF;
        if (tile + 1 < N_TOK / 16) {
            unsigned nxtOff = ((tile + 1) & 1) ? EB1_OFF : EB0_OFF;
            async_tile_load(zf + (size_t)(tile + 1) * 16 * CDIM, nxtOff, tid);
            wait_async_le8();
        } else {
            wait_async_zero();
        }
        __syncthreads();

        int n = tile * 16 + r;                      // this lane's token (column)
        float mn  = rowM[n];
        float is1 = 1.0f / rowS1[n];

        v8f acc = {0.f,0.f,0.f,0.f,0.f,0.f,0.f,0.f};
        const float* brow = lds + curOff / 4 + r * LSTR + 2 * hi;
        #pragma unroll
        for (int t = 0; t < 64; ++t) {
            v2f b; b.x = brow[4*t]; b.y = brow[4*t+1];
            acc = __builtin_amdgcn_wmma_f32_16x16x4_f32(false, aF[t], false, b,
                                                        (short)0, acc, false, false);
        }
        #pragma unroll
        for (int v = 0; v < 8; ++v)
            sumP[v] += __expf(200.0f * acc[v] - mn) * is1;   // exponent <= 0 always
        __syncthreads();
    }

    #pragma unroll
    for (int off = 1; off < 16; off <<= 1) {
        #pragma unroll
        for (int v = 0; v < 8; ++v) sumP[v] += __shfl_xor(sumP[v], off);
    }
    if (r == 0) {
        #pragma unroll
        for (int v = 0; v < 8; ++v)
            avgProbs[codeBase + wave * 16 + v + 8 * hi] = sumP[v] * (1.0f / (float)N_TOK);
    }
}

// ---------------------------------------------------------------------------
// Final scalars + idx (single block).
// ---------------------------------------------------------------------------
__global__ __launch_bounds__(256) void k_final(const float* __restrict__ rowS1,
                                               const float* __restrict__ rowS2,
                                               const float* __restrict__ rowBest,
                                               const int*   __restrict__ rowIdx,
                                               const float* __restrict__ avgProbs,
                                               const int*   __restrict__ qids,
                                               float* __restrict__ outScalars,
                                               float* __restrict__ outIdx) {
    __shared__ float red[256 * 3];
    __shared__ int   redI[256];
    int tid = threadIdx.x;
    float se = 0.f, vq = 0.f, ae = 0.f; int dead = 0;
    for (int n = tid; n < N_TOK; n += 256) {
        float s1 = rowS1[n];
        se += rowS2[n] / s1 - __logf(s1);           // sum_k p*logp per row
        int b = n >> 10, w = n & (WDIM - 1);
        if (w <= qids[b]) vq += 2.0f - 2.0f * rowBest[n];  // C*mean_C (zq-zt)^2
        outIdx[n] = (float)rowIdx[n];
    }
    for (int k = tid; k < N_EMB; k += 256) {
        float q = avgProbs[k];
        ae += q * __logf(q + 1e-5f);
    }
    for (int p = tid; p < HW; p += 256) {           // std<1e-5 <=> all batch equal
        int c0 = rowIdx[p], same = 1;
        for (int b = 1; b < BDIM; ++b) same &= (rowIdx[b * HW + p] == c0);
        dead += same;
    }
    red[tid] = se; red[256 + tid] = vq; red[512 + tid] = ae; redI[tid] = dead;
    __syncthreads();
    for (int s = 128; s > 0; s >>= 1) {
        if (tid < s) {
            red[tid]       += red[tid + s];
            red[256 + tid] += red[256 + tid + s];
            red[512 + tid] += red[512 + tid + s];
            redI[tid]      += redI[tid + s];
        }
        __syncthreads();
    }
    if (tid == 0) {
        float denom = 0.f;                           // mask is [B,1,W,1]: no H factor
        for (int b = 0; b < BDIM; ++b) denom += (float)(qids[b] + 1);
        float vql = red[256] / ((float)CDIM * denom);
        outScalars[0] = vql;                         // vq_loss
        outScalars[1] = 0.25f * vql;                 // commit_loss = BETA * same value
        outScalars[2] = 0.1f * (-(red[0]) / (float)N_TOK + red[512]); // entropy_loss
        outScalars[3] = (float)redI[0] / (float)HW;  // dead_code_rate
    }
}

// z_q output: gather normalized codes, transposed to [B,C,H,W]; coalesced writes.
__global__ __launch_bounds__(256) void k_zq(const float* __restrict__ embN,
                                            const int*   __restrict__ rowIdx,
                                            float* __restrict__ outZ) {
    int blk = blockIdx.x;                // b*256 + c
    int b = blk >> 8, c = blk & 255;
    const int* idxRow = rowIdx + b * HW;
    float* dst = outZ + blk * HW;
    #pragma unroll
    for (int j = 0; j < 4; ++j) {
        int hw = j * 256 + threadIdx.x;
        dst[hw] = embN[idxRow[hw] * CDIM + c];
    }
}

extern "C" void kernel_launch(void* const* d_in, const int* in_sizes, int n_in,
                              void* d_out, int out_size, void* d_ws, size_t ws_size,
                              hipStream_t stream) {
    const float* z    = (const float*)d_in[0];   // [16,256,32,32]
    const float* emb  = (const float*)d_in[1];   // [16384,256]
    const int*   qids = (const int*)d_in[2];     // [16,1]
    float* out = (float*)d_out;
    float* ws  = (float*)d_ws;

    float* zf       = ws;                        // 4194304
    float* embN     = zf + 4194304;              // 4194304
    float* rowM     = embN + 4194304;            // 16384
    float* rowS1    = rowM + N_TOK;
    float* rowS2    = rowS1 + N_TOK;
    float* rowBest  = rowS2 + N_TOK;
    int*   rowIdx   = (int*)(rowBest + N_TOK);
    float* avgProbs = (float*)(rowIdx + N_TOK);  // total ~32.4 MB

    k_normalize<<<4096, 256, 0, stream>>>(z, emb, zf, embN);

    k_phase1<<<N_TOK / 64, 128, PHASE_LDS_BYTES, stream>>>(zf, embN, rowM, rowS1, rowS2,
                                                           rowBest, rowIdx);
    k_phase2<<<N_EMB / 64, 128, PHASE_LDS_BYTES, stream>>>(zf, embN, rowM, rowS1, avgProbs);

    k_final<<<1, 256, 0, stream>>>(rowS1, rowS2, rowBest, rowIdx, avgProbs, qids,
                                   out + 4194304, out + 4194308);
    k_zq<<<BDIM * CDIM, 256, 0, stream>>>(embN, rowIdx, out);
}